// BackwardMaxPooling2D_13829794693491
// MI455X (gfx1250) — compile-verified
//
#include <hip/hip_runtime.h>

// ---------------------------------------------------------------------------
// BackwardMaxPooling2D (F_IBP) for MI455X / gfx1250.
//
// Outputs (concat, f32):
//   [0)              w_zero   : 33,554,432 zeros
//   [33554432)       b_out_u_ : 64
//   [33554496)       w_zero   : 33,554,432 zeros
//   [67108928)       b_out_l_ : 64
//
// Bandwidth-bound: ~268MB NT zero stores + ~67MB w reads + ~8MB pool reads
// => ~15us at 23.3 TB/s. Reduction math is done on the WMMA path
// (V_WMMA_F32_16X16X4_F32) using min(w,0) = w - relu(w):
//   out_u = sum relu(Wu)*(bu-bl) + Wu*bl + bias_u
//   out_l = sum relu(Wl)*(bl-bu) + Wl*bu + bias_l
// ---------------------------------------------------------------------------

typedef __attribute__((ext_vector_type(2))) float v2f;
typedef __attribute__((ext_vector_type(4))) float v4f;
typedef __attribute__((ext_vector_type(8))) float v8f;

#define N_POOL       131072            // pooled elements per batch (32*32*128)
#define N_OUT        32
#define WG_PER_BATCH 128
#define WAVES_PER_WG 8
#define P_PER_WAVE   (N_POOL / (WG_PER_BATCH * WAVES_PER_WG))   // 128

// --- 268MB of zeros: 128-bit non-temporal streaming stores ------------------
__global__ void __launch_bounds__(256) zero_fill_kernel(v4f* __restrict__ p, long n4) {
  v4f z = {0.f, 0.f, 0.f, 0.f};
  long i = (long)blockIdx.x * blockDim.x + threadIdx.x;
  long stride = (long)gridDim.x * blockDim.x;
  for (; i < n4; i += stride)
    __builtin_nontemporal_store(z, p + i);   // global_store_b128 ... nt
}

// --- 2x2/2 max pool of u_c and l_c (NHWC, B=2,H=W=64,C=128) -----------------
__global__ void __launch_bounds__(256) maxpool_kernel(const float* __restrict__ u_c,
                                                      const float* __restrict__ l_c,
                                                      float* __restrict__ bu,
                                                      float* __restrict__ bl) {
  int i = blockIdx.x * blockDim.x + threadIdx.x;   // (b,ph,pw,c) flattened
  if (i >= 2 * N_POOL) return;
  int c  = i & 127;
  int pw = (i >> 7) & 31;
  int ph = (i >> 12) & 31;
  int b  = i >> 17;
  long base = ((long)((b * 64 + 2 * ph) * 64 + 2 * pw)) * 128 + c;
  float a0 = u_c[base], a1 = u_c[base + 128], a2 = u_c[base + 8192], a3 = u_c[base + 8320];
  bu[i] = fmaxf(fmaxf(a0, a1), fmaxf(a2, a3));
  float c0 = l_c[base], c1 = l_c[base + 128], c2 = l_c[base + 8192], c3 = l_c[base + 8320];
  bl[i] = fmaxf(fmaxf(c0, c1), fmaxf(c2, c3));
}

// --- Interval GEMV over w via V_WMMA_F32_16X16X4_F32 ------------------------
// A (16x4 f32): row M=0 holds the b-vector slice (lane 0: K=0/1, lane 16: K=2/3).
// B (4x16 f32): VGPR0 = rows {0,2}, VGPR1 = rows {1,3}, columns striped on lanes.
// Accumulator D row M=0 = VGPR0 lanes 0..15 -> per-wave partial sums.
__global__ void __launch_bounds__(256) wmma_reduce_kernel(
    const float* __restrict__ wu, const float* __restrict__ wl,
    const float* __restrict__ bu, const float* __restrict__ bl,
    float* __restrict__ partial) {
  __shared__ float red[4 * WAVES_PER_WG * 16];
  const int tid  = threadIdx.x;
  const int lane = tid & 31;
  const int wave = tid >> 5;
  const int half = lane >> 4;      // 0: K rows 0/1, cols 0..15 ; 1: K rows 2/3
  const int l15  = lane & 15;
  const int b    = blockIdx.x / WG_PER_BATCH;
  const int wg   = blockIdx.x % WG_PER_BATCH;
  const long pool_base = (long)b * N_POOL;
  const long w_base    = pool_base * N_OUT;
  const int  p_start   = (wg * WAVES_PER_WG + wave) * P_PER_WAVE;
  const int  k0        = half * 2;

  v8f acc_u_lo = {}; v8f acc_u_hi = {}; v8f acc_l_lo = {}; v8f acc_l_hi = {};

  for (int blk = 0; blk < P_PER_WAVE / 4; ++blk) {
    const int p0 = p_start + blk * 4;

    // A-matrix row 0: only lanes 0 and 16 carry data (per f32 A layout)
    float bu0 = 0.f, bu1 = 0.f, bl0 = 0.f, bl1 = 0.f;
    if (l15 == 0) {
      bu0 = bu[pool_base + p0 + k0];
      bu1 = bu[pool_base + p0 + k0 + 1];
      bl0 = bl[pool_base + p0 + k0];
      bl1 = bl[pool_base + p0 + k0 + 1];
    }
    v2f A_du = {bu0 - bl0, bu1 - bl1};   // u-side: relu(Wu) * (bu - bl)
    v2f A_sl = {bl0, bl1};               // u-side: Wu * bl
    v2f A_dl = {bl0 - bu0, bl1 - bu1};   // l-side: relu(Wl) * (bl - bu)
    v2f A_su = {bu0, bu1};               // l-side: Wl * bu

    // B tiles: w[b, p0+k, o], rows contiguous by o (coalesced b32 loads)
    const long r = w_base + (long)(p0 + k0) * N_OUT + l15;
    __builtin_prefetch(&wu[r + 16 * N_OUT], 0, 3);   // global_prefetch_b8
    __builtin_prefetch(&wl[r + 16 * N_OUT], 0, 3);
    v2f Bu_lo = {wu[r],      wu[r + N_OUT]};
    v2f Bu_hi = {wu[r + 16], wu[r + N_OUT + 16]};
    v2f Bl_lo = {wl[r],      wl[r + N_OUT]};
    v2f Bl_hi = {wl[r + 16], wl[r + N_OUT + 16]};
    v2f Ru_lo = {fmaxf(Bu_lo.x, 0.f), fmaxf(Bu_lo.y, 0.f)};
    v2f Ru_hi = {fmaxf(Bu_hi.x, 0.f), fmaxf(Bu_hi.y, 0.f)};
    v2f Rl_lo = {fmaxf(Bl_lo.x, 0.f), fmaxf(Bl_lo.y, 0.f)};
    v2f Rl_hi = {fmaxf(Bl_hi.x, 0.f), fmaxf(Bl_hi.y, 0.f)};

    acc_u_lo = __builtin_amdgcn_wmma_f32_16x16x4_f32(false, A_du, false, Ru_lo, (short)0, acc_u_lo, false, false);
    acc_u_lo = __builtin_amdgcn_wmma_f32_16x16x4_f32(false, A_sl, false, Bu_lo, (short)0, acc_u_lo, false, false);
    acc_u_hi = __builtin_amdgcn_wmma_f32_16x16x4_f32(false, A_du, false, Ru_hi, (short)0, acc_u_hi, false, false);
    acc_u_hi = __builtin_amdgcn_wmma_f32_16x16x4_f32(false, A_sl, false, Bu_hi, (short)0, acc_u_hi, false, false);
    acc_l_lo = __builtin_amdgcn_wmma_f32_16x16x4_f32(false, A_dl, false, Rl_lo, (short)0, acc_l_lo, false, false);
    acc_l_lo = __builtin_amdgcn_wmma_f32_16x16x4_f32(false, A_su, false, Bl_lo, (short)0, acc_l_lo, false, false);
    acc_l_hi = __builtin_amdgcn_wmma_f32_16x16x4_f32(false, A_dl, false, Rl_hi, (short)0, acc_l_hi, false, false);
    acc_l_hi = __builtin_amdgcn_wmma_f32_16x16x4_f32(false, A_su, false, Bl_hi, (short)0, acc_l_hi, false, false);
  }

  // D row M=0 lives in acc[0], lanes 0..15 (N = lane). Cross-wave LDS reduce.
  if (lane < 16) {
    red[(0 * WAVES_PER_WG + wave) * 16 + lane] = acc_u_lo[0];
    red[(1 * WAVES_PER_WG + wave) * 16 + lane] = acc_u_hi[0];
    red[(2 * WAVES_PER_WG + wave) * 16 + lane] = acc_l_lo[0];
    red[(3 * WAVES_PER_WG + wave) * 16 + lane] = acc_l_hi[0];
  }
  __syncthreads();
  if (tid < 64) {
    int acc = tid >> 4;            // 0:u_lo 1:u_hi 2:l_lo 3:l_hi
    int o   = tid & 15;
    float s = 0.f;
    for (int w = 0; w < WAVES_PER_WG; ++w)
      s += red[(acc * WAVES_PER_WG + w) * 16 + o];
    partial[((long)blockIdx.x * 4 + acc) * 16 + o] = s;   // deterministic
  }
}

// --- Deterministic final reduction + bias (128 outputs total) ---------------
__global__ void finalize_kernel(const float* __restrict__ partial,
                                const float* __restrict__ bias_u,
                                const float* __restrict__ bias_l,
                                float* __restrict__ out_u,
                                float* __restrict__ out_l) {
  int t   = threadIdx.x;           // 0..127
  int b   = t >> 6;
  int acc = (t >> 4) & 3;
  int o   = t & 15;
  float s = 0.f;
  for (int wg = 0; wg < WG_PER_BATCH; ++wg)
    s += partial[((long)((b * WG_PER_BATCH + wg) * 4 + acc)) * 16 + o];
  int col = ((acc & 1) << 4) + o;
  if (acc < 2)
    out_u[b * 32 + col] = bias_u[b * 32 + col] + s;
  else
    out_l[b * 32 + col] = bias_l[b * 32 + col] + s;
}

extern "C" void kernel_launch(void* const* d_in, const int* in_sizes, int n_in,
                              void* d_out, int out_size, void* d_ws, size_t ws_size,
                              hipStream_t stream) {
  // inputs: 0:y 1:x_0 2:u_c 3:l_c 4:w_out_u 5:b_out_u 6:w_out_l 7:b_out_l
  const float* u_c    = (const float*)d_in[2];
  const float* l_c    = (const float*)d_in[3];
  const float* w_u    = (const float*)d_in[4];
  const float* bias_u = (const float*)d_in[5];
  const float* w_l    = (const float*)d_in[6];
  const float* bias_l = (const float*)d_in[7];
  float* out = (float*)d_out;

  const long N_ZERO = 33554432;                 // per w_zero block
  float* out_bu = out + N_ZERO;                 // b_out_u_ (64 floats)
  float* zero2  = out + N_ZERO + 64;            // second w_zero block
  float* out_bl = out + 2 * N_ZERO + 64;        // b_out_l_ (64 floats)

  // workspace: bu | bl | per-WG partials
  float* bu      = (float*)d_ws;
  float* bl      = bu + 2 * N_POOL;             // 262144 floats each
  float* partial = bl + 2 * N_POOL;             // 2*128*64 = 16384 floats

  zero_fill_kernel<<<2048, 256, 0, stream>>>((v4f*)out,   N_ZERO / 4);
  zero_fill_kernel<<<2048, 256, 0, stream>>>((v4f*)zero2, N_ZERO / 4);
  maxpool_kernel<<<(2 * N_POOL) / 256, 256, 0, stream>>>(u_c, l_c, bu, bl);
  wmma_reduce_kernel<<<2 * WG_PER_BATCH, 256, 0, stream>>>(w_u, w_l, bu, bl, partial);
  finalize_kernel<<<1, 128, 0, stream>>>(partial, bias_u, bias_l, out_bu, out_bl);
}